// DialogSeqAttnMatch_16209206575367
// MI455X (gfx1250) — compile-verified
//
#include <hip/hip_runtime.h>

typedef __attribute__((ext_vector_type(2))) float v2f;
typedef __attribute__((ext_vector_type(4))) float v4f;
typedef __attribute__((ext_vector_type(8))) float v8f;

#define Hd    128
#define Bn    32
#define L1n   512
#define NTOK  2048      // Bn * 64
#define LDSW  132       // padded LDS row stride (floats) for 128-wide tiles (bank skew 4)
#define PSTR  68        // padded LDS row stride for 64-wide P staging tile
#define TILEF (64 * LDSW)   // floats per 64x128 LDS tile (8448)

// D = A(16x4,f32) * B(4x16,f32) + C(16x16,f32)   — CDNA5 FP32 matrix pipe
static __device__ __forceinline__ v8f wmma_f32(v2f a, v2f b, v8f c) {
    return __builtin_amdgcn_wmma_f32_16x16x4_f32(false, a, false, b, (short)0, c, false, false);
}

static __device__ __forceinline__ float half16_max(float v) {
    v = fmaxf(v, __shfl_xor(v, 8, 32));
    v = fmaxf(v, __shfl_xor(v, 4, 32));
    v = fmaxf(v, __shfl_xor(v, 2, 32));
    v = fmaxf(v, __shfl_xor(v, 1, 32));
    return v;
}
static __device__ __forceinline__ float half16_sum(float v) {
    v += __shfl_xor(v, 8, 32);
    v += __shfl_xor(v, 4, 32);
    v += __shfl_xor(v, 2, 32);
    v += __shfl_xor(v, 1, 32);
    return v;
}

// ---------------------------------------------------------------------------
// CDNA5 async global->LDS staging (ASYNCcnt path), with safe fallback.
// Probe-derived parameter type: (vector_size(16) int AS1*, ... AS3*, imm, imm)
// ---------------------------------------------------------------------------
#if defined(__HIP_DEVICE_COMPILE__) && __has_builtin(__builtin_amdgcn_global_load_async_to_lds_b128)
#define HAVE_ASYNC_LDS 1
typedef int vs4i __attribute__((vector_size(16)));
typedef __attribute__((address_space(1))) vs4i g_v4i;   // global int4
typedef __attribute__((address_space(3))) vs4i l_v4i;   // LDS int4
#else
#define HAVE_ASYNC_LDS 0
#endif

static __device__ __forceinline__ void copy16_to_lds(const float* g, float* l) {
#if HAVE_ASYNC_LDS
    __builtin_amdgcn_global_load_async_to_lds_b128((g_v4i*)g, (l_v4i*)l, 0, 0);
#else
    *(v4f*)l = *(const v4f*)g;
#endif
}

static __device__ __forceinline__ void wait_async_tile(bool more_in_flight) {
#if HAVE_ASYNC_LDS
#if __has_builtin(__builtin_amdgcn_s_wait_asynccnt)
    if (more_in_flight) __builtin_amdgcn_s_wait_asynccnt(16);
    else                __builtin_amdgcn_s_wait_asynccnt(0);
#else
    if (more_in_flight) asm volatile("s_wait_asynccnt 0x10" ::: "memory");
    else                asm volatile("s_wait_asynccnt 0x0" ::: "memory");
#endif
#else
    (void)more_in_flight;
#endif
}

// ---------------------------------------------------------------------------
// Kernel 1: P = relu(X @ W^T + bias). 128 rows per block, 8 waves x 16 rows.
// mode 0: X = xd_emb rows.  mode 1: X = gathered dialog rows (xq|xa), and the
// gathered rows are also written contiguously to dlg_out for the V matmul.
// ---------------------------------------------------------------------------
__global__ __launch_bounds__(256, 1) void proj_relu_gemm(
    const float* __restrict__ xd,
    const float* __restrict__ xq,
    const float* __restrict__ xa,
    const float* __restrict__ W,      // (H,H) row-major
    const float* __restrict__ bias,   // (H,)
    float* __restrict__ proj_out,     // (rows,H)
    float* __restrict__ dlg_out,      // (NTOK,H), used only in mode 1
    int mode)
{
    extern __shared__ float sW[];     // Hd * LDSW floats

    const int tid   = threadIdx.x;
    const int lane  = tid & 31;
    const int wave  = tid >> 5;
    const int lrow  = lane & 15;
    const int halfk = (lane >> 4) * 2;    // 0 or 2: upper half-wave holds K+2,K+3

    // stage W into LDS (row-major, padded stride)
#pragma unroll
    for (int i = 0; i < 16; ++i) {
        int lin = tid + i * 256;          // float4 units, 4096 total = 128 rows * 32
        int r   = lin >> 5;
        int c4  = lin & 31;
        *(v4f*)(&sW[r * LDSW + c4 * 4]) = *(const v4f*)(W + r * Hd + c4 * 4);
    }

    // per-lane source row for the A stripe of this wave
    const int rowg = blockIdx.x * 128 + wave * 16 + lrow;
    const float* arow;
    if (mode == 0) {
        arow = xd + (size_t)rowg * Hd;
    } else {
        int bidx = rowg >> 6, pos = rowg & 63;
        arow = (pos < 32) ? (xq + (size_t)(bidx * 32 + pos) * Hd)
                          : (xa + (size_t)(bidx * 32 + (pos - 32)) * Hd);
    }

    // A fragments: lane<16 holds (k,k+1) of row lrow, lane>=16 holds (k+2,k+3)
    v2f af[32];
#pragma unroll
    for (int i = 0; i < 32; ++i)
        af[i] = *(const v2f*)(arow + i * 4 + halfk);

    if (mode == 1) {
        // materialize gathered dialog rows contiguously
#pragma unroll
        for (int i = 0; i < 16; ++i) {
            int lin = tid + i * 256;      // float4 units
            int r   = lin >> 5;
            int c4  = lin & 31;
            int g   = blockIdx.x * 128 + r;
            int bidx = g >> 6, pos = g & 63;
            const float* src = (pos < 32) ? (xq + (size_t)(bidx * 32 + pos) * Hd)
                                          : (xa + (size_t)(bidx * 32 + (pos - 32)) * Hd);
            *(v4f*)(dlg_out + (size_t)g * Hd + c4 * 4) = *(const v4f*)(src + c4 * 4);
        }
    }

    __syncthreads();

    v8f acc[8] = {};
#pragma unroll
    for (int i = 0; i < 32; ++i) {
        v2f a = af[i];
#pragma unroll
        for (int nt = 0; nt < 8; ++nt) {
            // B(k,j) = W[j,k]; j = nt*16 + lane, consecutive k -> one b64 read
            const float* bp = &sW[(nt * 16 + lrow) * LDSW + i * 4 + halfk];
            acc[nt] = wmma_f32(a, *(const v2f*)bp, acc[nt]);
        }
    }

    // epilogue: bias + relu + store (D layout: vgpr r -> rows r / r+8)
    const int rbase = blockIdx.x * 128 + wave * 16 + ((lane >> 4) << 3);
#pragma unroll
    for (int nt = 0; nt < 8; ++nt) {
        int col  = nt * 16 + lrow;
        float bv = bias[col];
#pragma unroll
        for (int r = 0; r < 8; ++r) {
            float v = acc[nt][r] + bv;
            proj_out[(size_t)(rbase + r) * Hd + col] = (v > 0.0f) ? v : 0.0f;
        }
    }
}

// ---------------------------------------------------------------------------
// Per-tile attention body: S = Xp@Yp^T, online softmax, O += P@V.
// ---------------------------------------------------------------------------
static __device__ __forceinline__ void attn_tile_body(
    const float* sYt, const float* sVt, float* sPw,
    const v2f* af, v8f* o, float* mrow, float* lsum,
    float tb, const bool* masked,
    int lrow, int halfk, int uphalf)
{
    // ---- S = Xp @ Yp^T : 16 x 64 (4 n-tiles), K=128 ----
    v8f s[4] = {};
#pragma unroll
    for (int i = 0; i < 32; ++i) {
        v2f a = af[i];
#pragma unroll
        for (int nt = 0; nt < 4; ++nt) {
            const float* bp = &sYt[(nt * 16 + lrow) * LDSW + i * 4 + halfk];
            s[nt] = wmma_f32(a, *(const v2f*)bp, s[nt]);
        }
    }

    // ---- padding mask + recency bias + online softmax ----
    float mx[8];
#pragma unroll
    for (int r = 0; r < 8; ++r) mx[r] = -3.0e38f;
#pragma unroll
    for (int nt = 0; nt < 4; ++nt) {
#pragma unroll
        for (int r = 0; r < 8; ++r) {
            float v = masked[nt] ? -3.0e38f : (s[nt][r] + tb);
            s[nt][r] = v;
            mx[r] = fmaxf(mx[r], v);
        }
    }

#pragma unroll
    for (int r = 0; r < 8; ++r) {
        float rowmax = half16_max(mx[r]);
        float mnew   = fmaxf(mrow[r], rowmax);
        float scale  = __expf(mrow[r] - mnew);
        mrow[r] = mnew;
        lsum[r] *= scale;
#pragma unroll
        for (int nt = 0; nt < 8; ++nt) o[nt][r] *= scale;
        float psum = 0.0f;
#pragma unroll
        for (int nt = 0; nt < 4; ++nt) {
            float p = masked[nt] ? 0.0f : __expf(s[nt][r] - mnew);
            s[nt][r] = p;
            psum += p;
        }
        lsum[r] += half16_sum(psum);
    }

    // stage P (D layout) to per-wave LDS scratch for A-layout reads
#pragma unroll
    for (int nt = 0; nt < 4; ++nt)
#pragma unroll
        for (int r = 0; r < 8; ++r)
            sPw[(r + uphalf) * PSTR + nt * 16 + lrow] = s[nt][r];

    // ---- O += P @ V : 16x64 * 64x128, K=64 ----
#pragma unroll
    for (int i = 0; i < 16; ++i) {
        v2f pa = *(const v2f*)(&sPw[lrow * PSTR + i * 4 + halfk]);
#pragma unroll
        for (int nt = 0; nt < 8; ++nt) {
            int col = nt * 16 + lrow;
            v2f vb;
            vb.x = sVt[(i * 4 + halfk) * LDSW + col];
            vb.y = sVt[(i * 4 + halfk + 1) * LDSW + col];
            o[nt] = wmma_f32(pa, vb, o[nt]);
        }
    }
}

// ---------------------------------------------------------------------------
// Kernel 2: fused flash attention over timestep tiles, double-buffered with
// async global->LDS staging. grid = (L1/128 row-tiles, B). Only timesteps
// t < b are unmasked so the K loop runs t = 0..b-1; the recency bias is a
// scalar per tile. Batch 0 -> zeros (zero_first).
// LDS layout: [buf0: sY|sV][buf1: sY|sV][sP]  (offsets computed at runtime;
// no pointer arrays — const-promoted LDS addrspacecasts break the linker).
// ---------------------------------------------------------------------------
__global__ __launch_bounds__(256, 1) void dialog_attn(
    const float* __restrict__ x_proj,          // (B*L1, H)
    const float* __restrict__ y_proj,          // (NTOK, H)
    const float* __restrict__ dlg,             // (NTOK, H)
    const float* __restrict__ rweight,         // (1,)
    const unsigned char* __restrict__ xq_mask, // (B,32) bool
    const unsigned char* __restrict__ xa_mask, // (B,32) bool
    float* __restrict__ out)                   // (B*L1, H)
{
    extern __shared__ float smem[];

    const int b   = blockIdx.y;
    const int tid = threadIdx.x;

    if (b == 0) {  // zero_first: whole batch-0 output is zero
        size_t base = (size_t)blockIdx.x * 128 * Hd;
#pragma unroll
        for (int i = 0; i < 16; ++i) {
            size_t lin = (size_t)tid + (size_t)i * 256;   // float4 units
            v4f z = {};
            *(v4f*)(out + base + lin * 4) = z;
        }
        return;
    }

    const int lane   = tid & 31;
    const int wave   = tid >> 5;
    const int lrow   = lane & 15;
    const int halfk  = (lane >> 4) * 2;
    const int uphalf = (lane >> 4) << 3;

    // A fragments: 16 rows of x_proj owned by this wave, kept in registers
    const float* arow =
        x_proj + ((size_t)b * L1n + blockIdx.x * 128 + wave * 16 + lrow) * Hd;
    v2f af[32];
#pragma unroll
    for (int i = 0; i < 32; ++i)
        af[i] = *(const v2f*)(arow + i * 4 + halfk);

    const float rw = rweight[0];

    float mrow[8], lsum[8];
#pragma unroll
    for (int r = 0; r < 8; ++r) { mrow[r] = -3.0e38f; lsum[r] = 0.0f; }
    v8f o[8] = {};

    float* sPw = smem + 4 * TILEF + wave * 16 * PSTR;

    // stage tile t into buffer bufsel: 16 b128 async fills per thread
    // (= 16 ASYNCcnt ticks per wave; <=32 in flight with double buffering)
#define STAGE_TILE(t_, bufsel_)                                                  \
    {                                                                            \
        float* sY_ = smem + (bufsel_) * (2 * TILEF);                             \
        float* sV_ = sY_ + TILEF;                                                \
        _Pragma("unroll")                                                        \
        for (int i_ = 0; i_ < 8; ++i_) {                                         \
            int lin_ = tid + i_ * 256;   /* float4 units, 2048 = 64 rows * 32 */ \
            int r_   = lin_ >> 5;                                                \
            int c4_  = lin_ & 31;                                                \
            copy16_to_lds(y_proj + ((size_t)(t_) * 64 + r_) * Hd + c4_ * 4,      \
                          &sY_[r_ * LDSW + c4_ * 4]);                            \
            copy16_to_lds(dlg + ((size_t)(t_) * 64 + r_) * Hd + c4_ * 4,         \
                          &sV_[r_ * LDSW + c4_ * 4]);                            \
        }                                                                        \
    }

    STAGE_TILE(0, 0);

    for (int t = 0; t < b; ++t) {
        const int cur   = t & 1;
        const bool more = (t + 1 < b);
        if (more) STAGE_TILE(t + 1, cur ^ 1);

        // retire exactly this tile's 16 async fills (next tile's 16 may remain)
        wait_async_tile(more);
        __syncthreads();

        const float tb = rw * (float)(b - t);
        bool masked[4];
#pragma unroll
        for (int nt = 0; nt < 4; ++nt) {
            int pos = nt * 16 + lrow;     // token position within timestep t (0..63)
            unsigned char mb = (pos < 32) ? xq_mask[t * 32 + pos]
                                          : xa_mask[t * 32 + (pos - 32)];
            masked[nt] = (mb != 0);
        }

        float* sYc = smem + cur * (2 * TILEF);
        float* sVc = sYc + TILEF;
        attn_tile_body(sYc, sVc, sPw, af, o, mrow, lsum,
                       tb, masked, lrow, halfk, uphalf);
        __syncthreads();
    }

    // epilogue: out = O / l
    const size_t row0 = (size_t)b * L1n + blockIdx.x * 128 + wave * 16;
#pragma unroll
    for (int r = 0; r < 8; ++r) {
        float inv = (lsum[r] > 0.0f) ? (1.0f / lsum[r]) : 0.0f;
#pragma unroll
        for (int nt = 0; nt < 8; ++nt) {
            int col = nt * 16 + lrow;
            out[(row0 + r + uphalf) * Hd + col] = o[nt][r] * inv;
        }
    }
#undef STAGE_TILE
}

// ---------------------------------------------------------------------------
extern "C" void kernel_launch(void* const* d_in, const int* in_sizes, int n_in,
                              void* d_out, int out_size, void* d_ws, size_t ws_size,
                              hipStream_t stream) {
    const float* xd      = (const float*)d_in[0];   // (32,512,128)
    const float* xq      = (const float*)d_in[1];   // (32,32,128)
    const float* xa      = (const float*)d_in[2];   // (32,32,128)
    const float* W       = (const float*)d_in[3];   // (128,128)
    const float* bias    = (const float*)d_in[4];   // (128,)
    const float* rweight = (const float*)d_in[5];   // (1,)
    const unsigned char* xq_mask = (const unsigned char*)d_in[6];  // (32,32) bool
    const unsigned char* xa_mask = (const unsigned char*)d_in[7];  // (32,32) bool
    float* out = (float*)d_out;

    // workspace layout (floats): x_proj | y_proj | dlg  (10 MB total)
    float* x_proj = (float*)d_ws;
    float* y_proj = x_proj + (size_t)Bn * L1n * Hd;     // +16384*128
    float* dlg    = y_proj + (size_t)NTOK * Hd;         // +2048*128

    const size_t smemA = (size_t)Hd * LDSW * sizeof(float);            // ~67.6 KB
    proj_relu_gemm<<<dim3(Bn * L1n / 128), 256, smemA, stream>>>(
        xd, xq, xa, W, bias, x_proj, dlg, 0);
    proj_relu_gemm<<<dim3(NTOK / 128), 256, smemA, stream>>>(
        xd, xq, xa, W, bias, y_proj, dlg, 1);

    // double-buffered sY/sV + per-wave P staging: 4*8448 + 8704 floats ≈ 166 KB
    const size_t smemB =
        (size_t)(4 * TILEF + 8 * 16 * PSTR) * sizeof(float);
    dialog_attn<<<dim3(L1n / 128, Bn), 256, smemB, stream>>>(
        x_proj, y_proj, dlg, rweight, xq_mask, xa_mask, out);
}